// IcoELLGATUNet_34333968564535
// MI455X (gfx1250) — compile-verified
//
#include <hip/hip_runtime.h>
#include <math.h>

// ---------------------------------------------------------------------------
// IcoELLGAT U-Net forward for MI455X (gfx1250, wave32).
// Dense work (W @ X per GAT layer) runs through V_WMMA_F32_16X16X4_F32 with
// fp32 accumulation (numerics match the fp32 JAX reference). GEMM uses 2x2
// register blocking per wave: 4 accumulators, 4 WMMAs per 6 VMEM loads,
// every A/B fragment reused twice. Whole model ~24 GFLOP / ~200MB -> L2
// resident on the 192MB global L2; bound by issue rate, not HBM.
// ---------------------------------------------------------------------------

typedef float v2f __attribute__((ext_vector_type(2)));
typedef float v8f __attribute__((ext_vector_type(8)));

#define LRELU(x) ((x) > 0.0f ? (x) : 0.01f * (x))

#define WMMA_F32(A, B, C) \
    __builtin_amdgcn_wmma_f32_16x16x4_f32(false, (A), false, (B), (short)0, (C), false, false)

// Y(M,N) = W(M,K) @ X(K,N), row-major, feature-major activations.
// M % 64 == 0, K % 4 == 0, N % 64 == 0.
// Block = 128 threads = 4 waves in a 2x2 arrangement; block tile 64x64,
// wave tile 32x32 = four 16x16 fp32 WMMA accumulators.
__global__ __launch_bounds__(128) void gemm_wmma_f32_kernel(
    const float* __restrict__ W, const float* __restrict__ X,
    float* __restrict__ Y, int M, int K, int N)
{
    const int lane = threadIdx.x & 31;
    const int wave = threadIdx.x >> 5;
    const int wm   = wave & 1;          // wave row in 2x2
    const int wn   = wave >> 1;         // wave col in 2x2
    const int mW   = (blockIdx.y << 6) + (wm << 5); // wave's first M row
    const int nW   = (blockIdx.x << 6) + (wn << 5); // wave's first N col
    const int t    = lane & 15;         // row-in-tile (A) / col-in-tile (B)
    const int kh   = (lane >> 4) << 1;  // K sub-offset 0 or 2

    // A fragments: K-contiguous pairs -> single b64 load each
    const float* wrow0 = W + (size_t)(mW + t) * K + kh;
    const float* wrow1 = wrow0 + (size_t)16 * K;
    // B fragments: two N-strided b32 loads each
    const float* xcol0 = X + (size_t)kh * N + (nW + t);
    const float* xcol1 = xcol0 + 16;

    v8f acc00 = {}, acc01 = {}, acc10 = {}, acc11 = {};
    for (int k = 0; k < K; k += 4) {
        v2f a0 = *(const v2f*)(wrow0 + k);
        v2f a1 = *(const v2f*)(wrow1 + k);
        v2f b0, b1;
        b0.x = xcol0[(size_t)k * N];
        b0.y = xcol0[(size_t)(k + 1) * N];
        b1.x = xcol1[(size_t)k * N];
        b1.y = xcol1[(size_t)(k + 1) * N];
        acc00 = WMMA_F32(a0, b0, acc00);
        acc01 = WMMA_F32(a0, b1, acc01);
        acc10 = WMMA_F32(a1, b0, acc10);
        acc11 = WMMA_F32(a1, b1, acc11);
    }

    // C/D layout per 16x16 tile: VGPR i -> M = i + 8*(lane>=16), N = lane&15
    const int rowOff = (lane >> 4) << 3;
    {
        float* y = Y + (size_t)(mW + rowOff) * N + (nW + t);
#pragma unroll
        for (int i = 0; i < 8; ++i) { y[(size_t)i * N] = acc00[i]; y[(size_t)i * N + 16] = acc01[i]; }
    }
    {
        float* y = Y + (size_t)(mW + 16 + rowOff) * N + (nW + t);
#pragma unroll
        for (int i = 0; i < 8; ++i) { y[(size_t)i * N] = acc10[i]; y[(size_t)i * N + 16] = acc11[i]; }
    }
}

// e[h,n] = sum_o G[h*O+o, n] * a[h,o]
__global__ void scores_kernel(const float* __restrict__ G,
                              const float* __restrict__ a,
                              float* __restrict__ E, int H, int O, int N)
{
    size_t id = (size_t)blockIdx.x * blockDim.x + threadIdx.x;
    if (id >= (size_t)H * N) return;
    int n = (int)(id % N);
    int h = (int)(id / N);
    const float* g  = G + (size_t)h * O * N + n;
    const float* av = a + (size_t)h * O;
    float s = 0.f;
    for (int o = 0; o < O; ++o) s += g[(size_t)o * N] * av[o];
    E[id] = s;
}

// alpha[h,e,n] = masked softmax_e( leaky_relu(eq[h,n] + ek[h, adj[n,e]]) )
__global__ void alpha_kernel(const float* __restrict__ eq,
                             const float* __restrict__ ek,
                             const int* __restrict__ adj,
                             float* __restrict__ alpha,
                             int E, int H, int Nq, int Nk)
{
    size_t id = (size_t)blockIdx.x * blockDim.x + threadIdx.x;
    if (id >= (size_t)H * Nq) return;
    int n = (int)(id % Nq);
    int h = (int)(id / Nq);
    const float eqv = eq[(size_t)h * Nq + n];
    const int*  ad  = adj + (size_t)n * E;

    float lg[7];
    float m = -1e30f;
#pragma unroll 7
    for (int e = 0; e < E; ++e) {
        int a  = ad[e];
        int ai = a < 0 ? 0 : a;
        float x = LRELU(eqv + ek[(size_t)h * Nk + ai]);
        if (a < 0) x = -1e30f;     // masked: exp(x-m) flushes to 0
        lg[e] = x;
        m = fmaxf(m, x);
    }
    float s = 0.f;
#pragma unroll 7
    for (int e = 0; e < E; ++e) {
        float w = expf(lg[e] - m);
        lg[e] = w;
        s += w;
    }
    float inv = 1.f / s;
    float* al = alpha + (size_t)h * E * Nq + n;
#pragma unroll 7
    for (int e = 0; e < E; ++e)
        al[(size_t)e * Nq] = lg[e] * inv;
}

// Y[row, n] = sum_e alpha[h,e,n] * G[row, adj[n,e]]   (optionally leaky_relu)
__global__ void aggregate_kernel(const float* __restrict__ G,
                                 const float* __restrict__ alpha,
                                 const int* __restrict__ adj,
                                 float* __restrict__ Y,
                                 int E, int H, int O, int Nq, int Nk, int relu)
{
    size_t id = (size_t)blockIdx.x * blockDim.x + threadIdx.x;
    size_t tot = (size_t)H * O * Nq;
    if (id >= tot) return;
    int n   = (int)(id % Nq);
    int row = (int)(id / Nq);
    int h   = row / O;
    const float* ga = G + (size_t)row * Nk;
    const float* al = alpha + (size_t)h * E * Nq + n;
    const int*   ad = adj + (size_t)n * E;
    float acc = 0.f;
#pragma unroll 7
    for (int e = 0; e < E; ++e) {
        int a  = ad[e];
        int ai = a < 0 ? 0 : a;
        acc += al[(size_t)e * Nq] * ga[ai];
    }
    if (relu) acc = LRELU(acc);
    Y[id] = acc;
}

// scatter_mean with arg = repeat(arange(Nc),4): exact /4 pooling.
__global__ void avg4_kernel(const float* __restrict__ in,
                            float* __restrict__ out, int F, int Nc)
{
    size_t id = (size_t)blockIdx.x * blockDim.x + threadIdx.x;
    if (id >= (size_t)F * Nc) return;
    int j = (int)(id % Nc);
    int f = (int)(id / Nc);
    const float* p = in + (size_t)f * ((size_t)Nc * 4) + (size_t)4 * j;
    out[id] = 0.25f * (p[0] + p[1] + p[2] + p[3]);
}

// Q[:, arg] with arg[i] = i//4: x4 upsample gather.
__global__ void up4_kernel(const float* __restrict__ in,
                           float* __restrict__ out, int F, int Nout)
{
    size_t id = (size_t)blockIdx.x * blockDim.x + threadIdx.x;
    if (id >= (size_t)F * Nout) return;
    int i = (int)(id % Nout);
    int f = (int)(id / Nout);
    out[id] = in[(size_t)f * (Nout >> 2) + (i >> 2)];
}

// softmax over R rows per column (final output, R=64).
__global__ void colsoftmax_kernel(const float* __restrict__ in,
                                  float* __restrict__ out, int R, int N)
{
    int n = blockIdx.x * blockDim.x + threadIdx.x;
    if (n >= N) return;
    float m = -1e30f;
    for (int r = 0; r < R; ++r) m = fmaxf(m, in[(size_t)r * N + n]);
    float s = 0.f;
    for (int r = 0; r < R; ++r) s += expf(in[(size_t)r * N + n] - m);
    float inv = 1.f / s;
    for (int r = 0; r < R; ++r)
        out[(size_t)r * N + n] = expf(in[(size_t)r * N + n] - m) * inv;
}

// ---------------------------------------------------------------------------
// Host-side orchestration
// ---------------------------------------------------------------------------

struct GatP { const float *W, *aq, *ak; int H, O, K; };

static inline void launch_gemm(hipStream_t st, const float* W, const float* X,
                               float* Y, int M, int K, int N)
{
    dim3 grid((unsigned)(N / 64), (unsigned)(M / 64), 1);
    gemm_wmma_f32_kernel<<<grid, 128, 0, st>>>(W, X, Y, M, K, N);
}

static inline unsigned nblk(size_t tot) { return (unsigned)((tot + 255) / 256); }

// same-graph GAT: Q == K
static void gat_same(hipStream_t st, const GatP& p, const int* adj, int E,
                     const float* X, int N, float* Y, int relu,
                     float* G, float* EQb, float* EKb, float* AL)
{
    const int HO = p.H * p.O;
    launch_gemm(st, p.W, X, G, HO, p.K, N);
    size_t hn = (size_t)p.H * N;
    scores_kernel<<<nblk(hn), 256, 0, st>>>(G, p.aq, EQb, p.H, p.O, N);
    scores_kernel<<<nblk(hn), 256, 0, st>>>(G, p.ak, EKb, p.H, p.O, N);
    alpha_kernel<<<nblk(hn), 256, 0, st>>>(EQb, EKb, adj, AL, E, p.H, N, N);
    size_t tot = (size_t)HO * N;
    aggregate_kernel<<<nblk(tot), 256, 0, st>>>(G, AL, adj, Y, E, p.H, p.O, N, N, relu);
}

// bipartite GAT: Q over Nq nodes, K over Nk nodes, adj holds K-side indices
static void gat_bip(hipStream_t st, const GatP& p,
                    const float* Xq, int Nq, const float* Xk, int Nk,
                    const int* adj, int E, float* Y, int relu,
                    float* G, float* GK, float* EQb, float* EKb, float* AL)
{
    const int HO = p.H * p.O;
    launch_gemm(st, p.W, Xq, G,  HO, p.K, Nq);
    launch_gemm(st, p.W, Xk, GK, HO, p.K, Nk);
    size_t hq = (size_t)p.H * Nq, hk = (size_t)p.H * Nk;
    scores_kernel<<<nblk(hq), 256, 0, st>>>(G,  p.aq, EQb, p.H, p.O, Nq);
    scores_kernel<<<nblk(hk), 256, 0, st>>>(GK, p.ak, EKb, p.H, p.O, Nk);
    alpha_kernel<<<nblk(hq), 256, 0, st>>>(EQb, EKb, adj, AL, E, p.H, Nq, Nk);
    size_t tot = (size_t)HO * Nq;
    aggregate_kernel<<<nblk(tot), 256, 0, st>>>(GK, AL, adj, Y, E, p.H, p.O, Nq, Nk, relu);
}

extern "C" void kernel_launch(void* const* d_in, const int* in_sizes, int n_in,
                              void* d_out, int out_size, void* d_ws, size_t ws_size,
                              hipStream_t stream)
{
    (void)in_sizes; (void)n_in; (void)out_size; (void)ws_size;

    // Param order: [c0l1, c0l2, in1, r01, c1l1, c1l2, r12, c2l1, c2l2,
    //               e2l1, e2l2, e1l1, e1l2, e0l1, e0l2, ro], each (W, aq, ak).
    static const int HOK[16][3] = {
        {4,32,16},{4,32,128},{4,64,16},{4,16,128},{4,64,320},{4,64,256},
        {4,32,256},{4,128,128},{4,128,512},{4,128,512},{4,128,512},
        {4,64,768},{4,64,256},{4,32,384},{4,32,128},{1,64,128}};
    GatP P[16];
    for (int i = 0; i < 16; ++i) {
        P[i].W  = (const float*)d_in[3 * i + 0];
        P[i].aq = (const float*)d_in[3 * i + 1];
        P[i].ak = (const float*)d_in[3 * i + 2];
        P[i].H = HOK[i][0]; P[i].O = HOK[i][1]; P[i].K = HOK[i][2];
    }
    const float* x0    = (const float*)d_in[48];
    const float* xin1  = (const float*)d_in[49];
    const int*   adj0  = (const int*)d_in[50];
    const int*   adj1  = (const int*)d_in[51];
    const int*   adj2  = (const int*)d_in[52];
    const int*   bip01 = (const int*)d_in[53];
    const int*   bip12 = (const int*)d_in[54];
    // arg01/arg12 (d_in[55..56]) are repeat(arange,4): handled structurally.

    const int N0 = 40960, N1 = 10240, N2 = 2560;

    // Workspace layout (floats)
    float* ws = (float*)d_ws;
    size_t o = 0;
    float* Z0  = ws + o; o += (size_t)128 * N0;     // skip @ level0
    float* Z1  = ws + o; o += (size_t)256 * N1;     // skip @ level1
    float* G   = ws + o; o += (size_t)12 << 20;     // gemm out (Q-side/unified)
    float* GK  = ws + o; o += (size_t)12 << 20;     // gemm out (K-side)
    float* XA  = ws + o; o += (size_t)16 << 20;     // activation arena A (incl. concats)
    float* YA  = ws + o; o += (size_t)16 << 20;     // activation arena B
    float* EQb = ws + o; o += (size_t)1 << 18;      // eq (H,N)
    float* EKb = ws + o; o += (size_t)1 << 18;      // ek (H,N)
    float* AL  = ws + o; o += (size_t)1 << 21;      // alpha (H,E,Nq)
    float* SM  = ws + o; o += (size_t)1 << 22;      // scatter-mean out

    // ---- contracting path -------------------------------------------------
    // c0: x0(16,N0) -> Z0(128,N0)
    gat_same(stream, P[0], adj0, 7, x0, N0, XA, 1, G, EQb, EKb, AL);
    gat_same(stream, P[1], adj0, 7, XA, N0, Z0, 1, G, EQb, EKb, AL);

    // r01: pooled Z0 (Q) vs Z0 (K) over bip01 -> cat rows [0,64) of XA(320,N1)
    avg4_kernel<<<nblk((size_t)128 * N1), 256, 0, stream>>>(Z0, SM, 128, N1);
    gat_bip(stream, P[3], SM, N1, Z0, N0, bip01, 6, XA, 1, G, GK, EQb, EKb, AL);
    // in1: pooled xin1 (Q) vs xin1 (K) -> cat rows [64,320)
    avg4_kernel<<<nblk((size_t)16 * N1), 256, 0, stream>>>(xin1, SM, 16, N1);
    gat_bip(stream, P[2], SM, N1, xin1, N0, bip01, 6, XA + (size_t)64 * N1, 1,
            G, GK, EQb, EKb, AL);

    // c1: XA(320,N1) -> Z1(256,N1)
    gat_same(stream, P[4], adj1, 7, XA, N1, YA, 1, G, EQb, EKb, AL);
    gat_same(stream, P[5], adj1, 7, YA, N1, Z1, 1, G, EQb, EKb, AL);

    // r12: pooled Z1 vs Z1 over bip12 -> XA(128,N2)
    avg4_kernel<<<nblk((size_t)256 * N2), 256, 0, stream>>>(Z1, SM, 256, N2);
    gat_bip(stream, P[6], SM, N2, Z1, N1, bip12, 6, XA, 1, G, GK, EQb, EKb, AL);

    // c2, e2 at level 2
    gat_same(stream, P[7],  adj2, 7, XA, N2, YA, 1, G, EQb, EKb, AL);
    gat_same(stream, P[8],  adj2, 7, YA, N2, XA, 1, G, EQb, EKb, AL);
    gat_same(stream, P[9],  adj2, 7, XA, N2, YA, 1, G, EQb, EKb, AL);
    gat_same(stream, P[10], adj2, 7, YA, N2, XA, 1, G, EQb, EKb, AL);

    // ---- expanding path ---------------------------------------------------
    // upsample to N1, concat Z1 -> YA(768,N1)
    up4_kernel<<<nblk((size_t)512 * N1), 256, 0, stream>>>(XA, YA, 512, N1);
    hipMemcpyAsync(YA + (size_t)512 * N1, Z1, (size_t)256 * N1 * sizeof(float),
                   hipMemcpyDeviceToDevice, stream);
    gat_same(stream, P[11], adj1, 7, YA, N1, XA, 1, G, EQb, EKb, AL);
    gat_same(stream, P[12], adj1, 7, XA, N1, YA, 1, G, EQb, EKb, AL);

    // upsample to N0, concat Z0 -> XA(384,N0)
    up4_kernel<<<nblk((size_t)256 * N0), 256, 0, stream>>>(YA, XA, 256, N0);
    hipMemcpyAsync(XA + (size_t)256 * N0, Z0, (size_t)128 * N0 * sizeof(float),
                   hipMemcpyDeviceToDevice, stream);
    gat_same(stream, P[13], adj0, 7, XA, N0, YA, 1, G, EQb, EKb, AL);
    gat_same(stream, P[14], adj0, 7, YA, N0, XA, 1, G, EQb, EKb, AL);

    // ro (heads=1, no relu) -> YA(64,N0); column softmax -> d_out
    gat_same(stream, P[15], adj0, 7, XA, N0, YA, 0, G, EQb, EKb, AL);
    colsoftmax_kernel<<<nblk((size_t)N0), 256, 0, stream>>>(YA, (float*)d_out, 64, N0);
}